// TensorFusion_34127810134237
// MI455X (gfx1250) — compile-verified
//
#include <hip/hip_runtime.h>

#define HID    96
#define BSZ    64
#define HP     97            // HID+1
#define JK     9409          // 97*97
#define JKP    9600          // padded: SPLITK * NCH * KC, zero-filled tail
#define ROWS   9312          // 96*97
#define KC     48            // K chunk staged in LDS (12 KB per buffer)
#define SPLITK 4
#define NCH    50            // chunks per K-split: (JKP/SPLITK)/KC
#define WAVES  6
#define TPB    (WAVES * 32)  // 192 threads
#define F4PC   12            // float4 per Mt row per chunk (KC/4)

typedef __attribute__((ext_vector_type(2))) float v2f;
typedef __attribute__((ext_vector_type(8))) float v8f;

// async-to-LDS builtin operand types: int4 vector pointers in AS1 / AS3
typedef int v4i_ __attribute__((vector_size(16)));
typedef v4i_ __attribute__((address_space(1))) as1_v4i;
typedef v4i_ __attribute__((address_space(3))) as3_v4i;

#if defined(__HIP_DEVICE_COMPILE__) && \
    __has_builtin(__builtin_amdgcn_global_load_async_to_lds_b128) && \
    __has_builtin(__builtin_amdgcn_s_wait_asynccnt)
#define HAS_ASYNC 1
#else
#define HAS_ASYNC 0
#endif

// ---------------------------------------------------------------------------
// Kernel 1: Mt[b*JKP + jk] = a'[b,j] * v'[b,k]   (TRANSPOSED: [b][jk], 0-pad)
// ---------------------------------------------------------------------------
__global__ void tfn_build_mt(const float* __restrict__ audio,
                             const float* __restrict__ video,
                             float* __restrict__ Mt) {
  int idx = blockIdx.x * blockDim.x + threadIdx.x;   // < 64*JKP
  int b  = idx / JKP;
  int jk = idx - b * JKP;
  float val = 0.f;
  if (jk < JK) {
    int j = jk / HP;
    int k = jk - j * HP;
    float aj = (j == 0) ? 1.f : audio[b * HID + (j - 1)];
    float vk = (k == 0) ? 1.f : video[b * HID + (k - 1)];
    val = aj * vk;
  }
  Mt[idx] = val;
}

// ---------------------------------------------------------------------------
// Kernel 2: split-K GEMM  Pp[s] = W1v[:, ks:ks+2400] @ Mt[ks:ks+2400, :]
//   grid = (97, SPLITK); block = 6 waves; wave w owns 16 rows x all 64 cols
// ---------------------------------------------------------------------------
__global__ __launch_bounds__(TPB) void tfn_gemm(const float* __restrict__ W1,
                                                const float* __restrict__ Mt,
                                                float* __restrict__ Pp) {
  __shared__ float lds[2][KC * BSZ];                 // [buf][n*KC + k], 2x12KB

  const int tid  = threadIdx.x;
  const int lane = tid & 31;
  const int ln   = lane & 15;                        // N / M sub-index
  const int kb   = (lane >> 4) << 1;                 // K phase: 0 or 2
  const int wave = tid >> 5;
  const int r0   = blockIdx.x * (WAVES * 16) + wave * 16;
  const int ks   = blockIdx.y * (JKP / SPLITK);      // this split's K base

  float* Pout = Pp + (size_t)blockIdx.y * ((size_t)ROWS * BSZ);

  // A-fragment row pointer (16x4 f32: lanes 0-15 K=0,1; lanes 16-31 K=2,3)
  const float* wp = W1 + (size_t)(r0 + ln) * (size_t)JK;

  const int t0   = tid;
  v8f acc[4] = {};

#if HAS_ASYNC
  // prologue: issue async fill of chunk 0 into buffer 0
#pragma unroll
  for (int it = 0; it < 4; ++it) {
    int t = t0 + it * TPB;
    int row = t / F4PC;
    int f4  = t - row * F4PC;
    const float* g = Mt + (size_t)row * JKP + ks + f4 * 4;
    const float* l = &lds[0][row * KC + f4 * 4];
    __builtin_amdgcn_global_load_async_to_lds_b128(
        (as1_v4i*)(uintptr_t)g,
        (as3_v4i*)(unsigned)(uintptr_t)l,
        0, 0);
  }
#endif

  for (int c = 0; c < NCH; ++c) {
    const int cur = c & 1;

#if HAS_ASYNC
    __builtin_amdgcn_s_wait_asynccnt(0);   // my fills for chunk c landed
    __syncthreads();                       // everyone's fills + prev compute done
    if (c + 1 < NCH) {                     // issue chunk c+1 under compute of c
#pragma unroll
      for (int it = 0; it < 4; ++it) {
        int t = t0 + it * TPB;
        int row = t / F4PC;
        int f4  = t - row * F4PC;
        const float* g = Mt + (size_t)row * JKP + ks + (c + 1) * KC + f4 * 4;
        const float* l = &lds[cur ^ 1][row * KC + f4 * 4];
        __builtin_amdgcn_global_load_async_to_lds_b128(
            (as1_v4i*)(uintptr_t)g,
            (as3_v4i*)(unsigned)(uintptr_t)l,
            0, 0);
      }
    }
#else
    __syncthreads();
#pragma unroll
    for (int it = 0; it < 4; ++it) {
      int t = t0 + it * TPB;
      int row = t / F4PC;
      int f4  = t - row * F4PC;
      *(float4*)&lds[cur][row * KC + f4 * 4] =
          *(const float4*)(Mt + (size_t)row * JKP + ks + c * KC + f4 * 4);
    }
    __syncthreads();
#endif

    const float* lb = lds[cur];
#pragma unroll 4
    for (int kk = 0; kk < KC; kk += 4) {
      // A fragment: stream W1, non-temporal (each element used once chip-wide).
      // Clamp column for the zero-padded K tail (B is 0 there).
      int kg  = ks + c * KC + kk + kb;
      int ka  = kg     < JK ? kg     : (JK - 1);
      int kbx = kg + 1 < JK ? kg + 1 : (JK - 1);
      v2f a;
      a.x = __builtin_nontemporal_load(wp + ka);
      a.y = __builtin_nontemporal_load(wp + kbx);

#pragma unroll
      for (int nt = 0; nt < 4; ++nt) {
        // B fragment 4x16 from [n][k] LDS layout: one contiguous b64 per lane
        v2f bq = *(const v2f*)&lb[(nt * 16 + ln) * KC + kk + kb];
        acc[nt] = __builtin_amdgcn_wmma_f32_16x16x4_f32(
            false, a, false, bq, (short)0, acc[nt], false, false);
      }
    }
  }

  // C/D layout: lane<16 -> M = v ; lane>=16 -> M = v+8 ; N = nt*16 + (lane&15)
  const int hi = (lane >> 4) * 8;
#pragma unroll
  for (int nt = 0; nt < 4; ++nt) {
#pragma unroll
    for (int v = 0; v < 8; ++v) {
      int row = r0 + v + hi;
      Pout[(size_t)row * BSZ + nt * 16 + ln] = acc[nt][v];
    }
  }
}

// ---------------------------------------------------------------------------
// Kernel 3a: h1[b][o] = relu(b1[o] + sum_i t'[b,i] * sum_s Pp[s][(o,i),b])
//   grid = 96 blocks (one per o) x 256 threads (64 b  x  4 i-slices)
// ---------------------------------------------------------------------------
__global__ __launch_bounds__(256) void tfn_h1(const float* __restrict__ text,
                                              const float* __restrict__ P,
                                              const float* __restrict__ b1,
                                              float* __restrict__ h1) {
  __shared__ float red[4][BSZ];
  const int o  = blockIdx.x;
  const int t  = threadIdx.x;
  const int b  = t & 63;
  const int q  = t >> 6;                             // i-slice 0..3
  const size_t PS = (size_t)ROWS * BSZ;              // partial stride

  float s = 0.f;
  for (int i = q; i < HP; i += 4) {
    float tv = (i == 0) ? 1.f : text[b * HID + (i - 1)];
    size_t off = (size_t)(o * HP + i) * BSZ + b;     // coalesced across b
    float p = (P[off] + P[off + PS]) + (P[off + 2 * PS] + P[off + 3 * PS]);
    s += tv * p;
  }
  red[q][b] = s;
  __syncthreads();
  if (q == 0) {
    float tot = (red[0][b] + red[1][b]) + (red[2][b] + red[3][b]) + b1[o];
    h1[b * HID + o] = fmaxf(tot, 0.f);
  }
}

// ---------------------------------------------------------------------------
// Kernel 3b: tiny MLP head.  1 x 64 threads, one per batch element.
// ---------------------------------------------------------------------------
__global__ void tfn_head(const float* __restrict__ h1,
                         const float* __restrict__ W2,
                         const float* __restrict__ b2,
                         const float* __restrict__ W3,
                         const float* __restrict__ b3,
                         float* __restrict__ out) {
  int b = threadIdx.x;
  if (b >= BSZ) return;

  float h2[HID / 2];
  for (int j = 0; j < HID / 2; ++j) {
    float s = b2[j];
    for (int o = 0; o < HID; ++o) s += W2[j * HID + o] * h1[b * HID + o];
    h2[j] = fmaxf(s, 0.f);
  }
  for (int c = 0; c < 3; ++c) {
    float s = b3[c];
    for (int j = 0; j < HID / 2; ++j) s += W3[c * (HID / 2) + j] * h2[j];
    out[b * 3 + c] = s;
  }
}

// ---------------------------------------------------------------------------
extern "C" void kernel_launch(void* const* d_in, const int* in_sizes, int n_in,
                              void* d_out, int out_size, void* d_ws, size_t ws_size,
                              hipStream_t stream) {
  const float* text  = (const float*)d_in[0];
  const float* audio = (const float*)d_in[1];
  const float* video = (const float*)d_in[2];
  const float* W1    = (const float*)d_in[3];
  const float* b1    = (const float*)d_in[4];
  const float* W2    = (const float*)d_in[5];
  const float* b2    = (const float*)d_in[6];
  const float* W3    = (const float*)d_in[7];
  const float* b3    = (const float*)d_in[8];
  float* out = (float*)d_out;

  float* Mt = (float*)d_ws;                           // [64 x JKP]          ~2.46 MB
  float* P  = Mt + (size_t)BSZ * JKP;                 // [SPLITK][9312 x 64] ~9.53 MB
  float* h1 = P + (size_t)SPLITK * ROWS * BSZ;        // [64 x 96]           24 KB

  tfn_build_mt<<<(BSZ * JKP) / 256, 256, 0, stream>>>(audio, video, Mt);
  tfn_gemm<<<dim3(ROWS / (WAVES * 16), SPLITK), TPB, 0, stream>>>(W1, Mt, P);
  tfn_h1<<<HID, 256, 0, stream>>>(text, P, b1, h1);
  tfn_head<<<1, 64, 0, stream>>>(h1, W2, b2, W3, b3, out);
}